// SimplifiedBiologicalNeuron_23983097381187
// MI455X (gfx1250) — compile-verified
//
#include <hip/hip_runtime.h>
#include <stdint.h>

// ---------------- problem constants (reference shapes) ----------------
constexpr int B = 32, T = 2048, H = 512;
constexpr int DSAMP = 4;
constexpr int NSTEP = T / DSAMP;          // 512 processed steps
constexpr float V_REST  = -70.0f;
constexpr float V_RESET = -75.0f;

// ---------------- tiling ----------------
constexpr int CW     = 128;               // h-chunk width == threads per block (4 waves)
constexpr int NBH    = H / CW;            // 4 h-chunks
constexpr int TROWS  = 32;                // timesteps per TDM tile
constexpr int NCHUNK = NSTEP / TROWS;     // 16 chunks
constexpr int TILE_BYTES = TROWS * CW * 4; // 16 KB per buffer

typedef __attribute__((ext_vector_type(4))) unsigned int u32x4;
typedef __attribute__((ext_vector_type(4))) int          i32x4;
typedef __attribute__((ext_vector_type(8))) int          i32x8;
typedef __attribute__((ext_vector_type(4))) float        f32x4;

#if defined(__gfx1250__) && __has_builtin(__builtin_amdgcn_tensor_load_to_lds) && __has_builtin(__builtin_amdgcn_s_wait_tensorcnt)
#define USE_TDM 1
#else
#define USE_TDM 0
#endif

// ---------------- counter-based RNG (PCG hash + Box-Muller) ----------------
__device__ __forceinline__ uint32_t pcg_hash(uint32_t x) {
    x = x * 747796405u + 2891336453u;
    uint32_t w = ((x >> ((x >> 28u) + 4u)) ^ x) * 277803737u;
    return (w >> 22u) ^ w;
}
__device__ __forceinline__ float u01(uint32_t r) {            // [0,1)
    return (float)(r >> 8) * 0x1.0p-24f;
}

#if USE_TDM
// Issue one TDM 2D tile load: TROWS rows x CW floats, row stride DSAMP*H elements.
// Executed by one wave only (instruction ignores EXEC; gate at wave granularity).
__device__ __forceinline__ void tdm_load_tile(const float* gsrc, uint32_t lds_byte_addr) {
    uint64_t ga = (uint64_t)(uintptr_t)gsrc;
    u32x4 g0;
    g0.x = 1u;                                             // count=1 valid descriptor
    g0.y = lds_byte_addr;                                  // LDS dest (byte addr)
    g0.z = (uint32_t)(ga & 0xFFFFFFFFu);                   // global_addr[31:0]
    g0.w = (uint32_t)((ga >> 32) & 0x1FFFFFFu)             // global_addr[56:32]
         | (2u << 30);                                     // type = 2 ("image")
    i32x8 g1;
    g1[0] = (int)(2u << 16);                               // data_size = 2 -> 4 bytes
    g1[1] = (int)((uint32_t)H << 16);                      // tensor_dim0[15:0]=H (abar=0)
    g1[2] = (int)(0x4000u << 16);                          // dim0 hi=0 | tensor_dim1 lo=16384
    g1[3] = (int)((uint32_t)CW << 16);                     // dim1 hi=0 | tile_dim0=CW
    g1[4] = (int)(uint32_t)TROWS;                          // tile_dim1=TROWS, tile_dim2=0
    g1[5] = (int)(uint32_t)(DSAMP * H);                    // tensor_dim0_stride = 4*H (row gather)
    g1[6] = 0;                                             // stride hi | dim1_stride lo
    g1[7] = 0;
    i32x4 z4 = {0, 0, 0, 0};                               // groups 2/3 unused (<=2D)
    i32x8 z8 = {0, 0, 0, 0, 0, 0, 0, 0};                   // 6-arg form (clang-23 headers)
    __builtin_amdgcn_tensor_load_to_lds(g0, g1, z4, z4, z8, 0);
}
#endif

// ---------------- zero-fill of the padded spike rows (rows NSTEP..T-1) ----------------
__global__ __launch_bounds__(256) void zero_pad_kernel(float* __restrict__ out) {
    const size_t per_b  = (size_t)(T - NSTEP) * H;        // floats of padding per batch
    const size_t total4 = (size_t)B * per_b / 4;
    for (size_t i = (size_t)blockIdx.x * blockDim.x + threadIdx.x; i < total4;
         i += (size_t)gridDim.x * blockDim.x) {
        size_t e = i * 4;
        size_t b = e / per_b;
        size_t r = e - b * per_b;
        f32x4 z = {0.0f, 0.0f, 0.0f, 0.0f};
        __builtin_nontemporal_store(z, (f32x4*)(out + ((size_t)b * T + NSTEP) * H + r));
    }
}

// ---------------- main LIF scan ----------------
__global__ __launch_bounds__(CW) void neuron_scan_kernel(
        const float* __restrict__ x,
        const float* __restrict__ p_base,
        const float* __restrict__ p_noise,
        const float* __restrict__ p_tbias,
        float* __restrict__ out) {
    __shared__ float smem[2][TROWS * CW];

    const int b    = blockIdx.x / NBH;
    const int h0   = (blockIdx.x % NBH) * CW;
    const int lane = threadIdx.x;

    const float basec = p_base[0];
    const float nstr  = p_noise[0];
    const float thr   = -55.0f + p_tbias[0];

    const float* gbase = x + (size_t)b * T * H + h0;      // column chunk of batch b
    const uint32_t gid = (uint32_t)(b * H + h0 + lane);

    float v = V_REST;

#if USE_TDM
    const uint32_t lds0 = (uint32_t)(uintptr_t)&smem[0][0];  // low 32 bits = LDS byte offset
    if (lane < 32) {                                      // wave 0 only issues DMA
        tdm_load_tile(gbase, lds0);
        tdm_load_tile(gbase + (size_t)TROWS * DSAMP * H, lds0 + TILE_BYTES);
    }
#endif

    for (int c = 0; c < NCHUNK; ++c) {
#if USE_TDM
        if (lane < 32) {
            if (c < NCHUNK - 1) __builtin_amdgcn_s_wait_tensorcnt(1);  // chunk c landed
            else                __builtin_amdgcn_s_wait_tensorcnt(0);
        }
        __syncthreads();
#else
        for (int r = 0; r < TROWS; ++r)
            smem[c & 1][r * CW + lane] =
                gbase[(size_t)(c * TROWS + r) * DSAMP * H + lane];
        __syncthreads();
#endif
        const float* tile = &smem[c & 1][0];
        #pragma unroll 4
        for (int r = 0; r < TROWS; ++r) {
            const int   s  = c * TROWS + r;
            const float xv = tile[r * CW + lane];

            // three independent uniforms per (neuron, step)
            const uint32_t seed = gid * 0x9E3779B9u ^ (uint32_t)s * 0x85EBCA77u ^ 0x2545F491u;
            const uint32_t r1 = pcg_hash(seed);
            const uint32_t r2 = pcg_hash(seed ^ 0x6A09E667u);
            const uint32_t r3 = pcg_hash(seed + 0xBB67AE85u);

            // Box-Muller normal
            const float u1  = fmaxf(u01(r1), 1.0e-9f);
            const float rad = __builtin_sqrtf(-2.0f * __logf(u1));
            const float nrm = rad * __cosf(6.28318530718f * u01(r2));

            const float I = xv + basec + nrm * nstr;
            v += 0.05f * (10.0f * I - (v - V_REST));      // (DT/TAU_MEM) = 0.05

            const bool spike = (v > thr) | (u01(r3) < 0.02f);
            __builtin_nontemporal_store(spike ? 1.0f : 0.0f,
                                        out + ((size_t)b * T + s) * H + h0 + lane);
            if (spike) v = V_RESET;
        }
        __syncthreads();                                  // buffer free before reissue
#if USE_TDM
        if (lane < 32 && (c + 2) < NCHUNK)
            tdm_load_tile(gbase + (size_t)(c + 2) * TROWS * DSAMP * H,
                          lds0 + (uint32_t)(c & 1) * TILE_BYTES);
#endif
    }

    // v_final appended after the spike tensor
    out[(size_t)B * T * H + (size_t)b * H + h0 + lane] = v;
}

extern "C" void kernel_launch(void* const* d_in, const int* in_sizes, int n_in,
                              void* d_out, int out_size, void* d_ws, size_t ws_size,
                              hipStream_t stream) {
    (void)in_sizes; (void)n_in; (void)d_ws; (void)ws_size; (void)out_size;
    const float* x      = (const float*)d_in[0];
    const float* basec  = (const float*)d_in[1];
    const float* nstr   = (const float*)d_in[2];
    const float* tbias  = (const float*)d_in[3];
    float*       out    = (float*)d_out;

    zero_pad_kernel<<<2048, 256, 0, stream>>>(out);
    neuron_scan_kernel<<<B * NBH, CW, 0, stream>>>(x, basec, nstr, tbias, out);
}